// HyperPatch_37177236914656
// MI455X (gfx1250) — compile-verified
//
#include <hip/hip_runtime.h>
#include <stdint.h>

// ---------------------------------------------------------------------------
// HyperPatch: boundary-aware patch sampling + gather for MI455X (gfx1250).
//
// Pipeline:
//   K1 hp_flags   : TDM tensor_load_to_lds stages 10x10x10 mask tiles into LDS
//                   (TENSORcnt-tracked async DMA), 8x8x8 voxels/block compute
//                   3x3x3 counts from LDS and emit per-voxel category flags.
//   K2 hp_sample  : deterministic hash-based rejection sampling (uniform over
//                   a flagged category, with fallback category, then uniform),
//                   matching the reference's primary->fallback semantics.
//   K3 hp_gather  : gather 3x3x3 x 8-channel patches with replicate padding
//                   (coordinate clamp), one output element per thread.
// ---------------------------------------------------------------------------

#define DVOL   64
#define S_VOL  (64 * 64 * 64)           // 262144 == 1<<18
#define NSAMP  32
#define BK     22                        // boundary_k = int(32*0.7)
#define ROWLEN 13824                     // 2 * ns * g * p3 = 2*32*8*27
#define NROWS  16                        // B*G
#define OUT_N  (NROWS * ROWLEN)          // 221184

typedef unsigned int u32x4 __attribute__((ext_vector_type(4)));
typedef int          i32x8 __attribute__((ext_vector_type(8)));
typedef int          i32x4 __attribute__((ext_vector_type(4)));

__device__ __forceinline__ unsigned hp_hash32(unsigned x) {
    x ^= x >> 16; x *= 0x7feb352dU;
    x ^= x >> 15; x *= 0x846ca68bU;
    x ^= x >> 16;
    return x;
}

__device__ __forceinline__ int hp_clamp63(int v) {
    return v < 0 ? 0 : (v > 63 ? 63 : v);
}

// flags bits: 1=pos(mask==1) 2=neg(mask==0) 4=pos_boundary 8=neg_boundary 16=valid interior
__global__ __launch_bounds__(512) void hp_flags_kernel(
    const int* __restrict__ mask, unsigned char* __restrict__ flags)
{
    __shared__ int tile[1000];                      // 10x10x10 window, 4000 B

    const int bid = blockIdx.x;
    const int b   = bid >> 9;                       // batch
    const int t   = bid & 511;                      // tile id within batch
    const int z0  = (t >> 6) * 8;
    const int y0  = ((t >> 3) & 7) * 8;
    const int x0  = (t & 7) * 8;
    // Shifted, fully in-bounds 10-wide window covering the tile + halo.
    int x0w = x0 - 1; x0w = x0w < 0 ? 0 : (x0w > 54 ? 54 : x0w);
    int y0w = y0 - 1; y0w = y0w < 0 ? 0 : (y0w > 54 ? 54 : y0w);
    int z0w = z0 - 1; z0w = z0w < 0 ? 0 : (z0w > 54 ? 54 : z0w);

    const unsigned tid = threadIdx.x;

#if __has_builtin(__builtin_amdgcn_tensor_load_to_lds)
    // --- Tensor Data Mover: DMA the 10x10x10 int32 tile into LDS -----------
    if (tid < 32u) {                                 // wave 0 only (uniform)
        const uint64_t gaddr = (uint64_t)(uintptr_t)mask +
            4ull * (uint64_t)(x0w + 64 * y0w + 4096 * z0w + 262144 * b);
        const unsigned lds_off = (unsigned)(uintptr_t)&tile[0];

        u32x4 g0;
        g0[0] = 1u;                                  // count=1, user descriptor
        g0[1] = lds_off;                             // lds_addr (bytes)
        g0[2] = (unsigned)gaddr;                     // global_addr[31:0]
        g0[3] = ((unsigned)(gaddr >> 32) & 0x01FFFFFFu) | (2u << 30); // addr[56:32], type=2

        i32x8 g1;
        g1[0] = (int)(2u << 16);                     // data_size=4B; wg_mask=0; no flags
        g1[1] = (int)(64u << 16);                    // tensor_dim0[15:0]=64
        g1[2] = (int)(64u << 16);                    // tensor_dim0[31:16]=0, tensor_dim1[15:0]=64
        g1[3] = (int)(10u << 16);                    // tensor_dim1[31:16]=0, tile_dim0=10
        g1[4] = (int)(10u | (10u << 16));            // tile_dim1=10, tile_dim2=10
        g1[5] = 64;                                  // tensor_dim0_stride[31:0]=64
        g1[6] = (int)(4096u << 16);                  // stride0[47:32]=0, stride1[15:0]=4096
        g1[7] = 0;                                   // stride1[47:16]=0

        i32x4 g2; g2[0] = 64; g2[1] = 0; g2[2] = 0; g2[3] = 0;  // tensor_dim2=64
        i32x4 g3; g3[0] = 0;  g3[1] = 0; g3[2] = 0; g3[3] = 0;
        i32x8 g4; g4[0] = 0;  g4[1] = 0; g4[2] = 0; g4[3] = 0;
                  g4[4] = 0;  g4[5] = 0; g4[6] = 0; g4[7] = 0;

        // amdgpu-toolchain (clang-23) 6-arg form:
        // (uint32x4 g0, int32x8 g1, int32x4 g2, int32x4 g3, int32x8 g4, i32 cpol)
        __builtin_amdgcn_tensor_load_to_lds(g0, g1, g2, g3, g4, 0);
#if __has_builtin(__builtin_amdgcn_s_wait_tensorcnt)
        __builtin_amdgcn_s_wait_tensorcnt(0);
#else
        asm volatile("s_wait_tensorcnt 0" ::: "memory");
#endif
    }
#else
    // Fallback: cooperative per-lane tile load.
    for (int i = (int)tid; i < 1000; i += 512) {
        const int dz = i / 100, rem = i - dz * 100;
        const int dy = rem / 10, dx = rem - dy * 10;
        tile[i] = mask[(size_t)b * S_VOL +
                       (size_t)(z0w + dz) * 4096 + (y0w + dy) * 64 + (x0w + dx)];
    }
#endif
    __syncthreads();

    const int lz = (int)(tid >> 6), ly = (int)((tid >> 3) & 7), lx = (int)(tid & 7);
    const int z = z0 + lz, y = y0 + ly, x = x0 + lx;

    const int m = tile[(z - z0w) * 100 + (y - y0w) * 10 + (x - x0w)];

    int cnt = 0;
#pragma unroll
    for (int dz = -1; dz <= 1; ++dz)
#pragma unroll
        for (int dy = -1; dy <= 1; ++dy)
#pragma unroll
            for (int dx = -1; dx <= 1; ++dx) {
                const int nz = z + dz, ny = y + dy, nx = x + dx;
                if ((unsigned)nz < 64u && (unsigned)ny < 64u && (unsigned)nx < 64u)
                    cnt += tile[(nz - z0w) * 100 + (ny - y0w) * 10 + (nx - x0w)];
            }

    unsigned f = (m == 1) ? 1u : 2u;
    if (m == 1 && cnt < 27) f |= 4u;
    if (m == 0 && cnt > 0)  f |= 8u;
    if ((unsigned)(z - 1) < 62u && (unsigned)(y - 1) < 62u && (unsigned)(x - 1) < 62u)
        f |= 16u;

    flags[(size_t)b * S_VOL + (size_t)z * 4096 + y * 64 + x] = (unsigned char)f;
}

// One block per (b, G) row: 64 samples (32 pos + 32 neg), rejection sampling.
__global__ __launch_bounds__(64) void hp_sample_kernel(
    const unsigned char* __restrict__ flags, int* __restrict__ centers)
{
    const int row  = blockIdx.x;          // b*8 + gidx
    const int b    = row >> 3;
    const unsigned tid  = threadIdx.x;    // 0..63
    const int kind = (int)(tid >> 5);     // 0 = pos, 1 = neg
    const int j    = (int)(tid & 31);

    unsigned prim, fb;
    if (kind == 0) { prim = (j < BK) ? 4u : 1u;  fb = (j < BK) ? 1u : 16u; }
    else           { prim = (j < BK) ? 8u : 2u;  fb = (j < BK) ? 2u : 16u; }

    const unsigned char* fl = flags + (size_t)b * S_VOL;
    const unsigned seed =
        (((unsigned)row * 2u + (unsigned)kind) * 32u + (unsigned)j) * 0x9E3779B9u + 0x5bd1e995u;

    int cand = 0;
    bool found = false;
    for (int t = 0; t < 64 && !found; ++t) {            // primary category
        cand = (int)(hp_hash32(seed + (unsigned)t * 0x6C8E9CF5u) & 0x3FFFFu);
        if (fl[cand] & prim) found = true;
    }
    for (int t = 64; t < 128 && !found; ++t) {          // fallback category
        cand = (int)(hp_hash32(seed + (unsigned)t * 0x6C8E9CF5u) & 0x3FFFFu);
        if (fl[cand] & fb) found = true;
    }
    // else: uniform over all voxels (cand as-is)

    centers[row * 64 + (int)tid] = cand;
}

// One output element per thread: out[(b,g), s*216 + c*27 + o]
__global__ __launch_bounds__(256) void hp_gather_kernel(
    const float* __restrict__ fmap, const int* __restrict__ centers,
    float* __restrict__ out)
{
    const int i = blockIdx.x * 256 + (int)threadIdx.x;
    if (i >= OUT_N) return;

    const int row = i / ROWLEN;
    const int off = i - row * ROWLEN;
    const int b    = row >> 3;
    const int gidx = row & 7;

    const int o  = off % 27;
    const int t2 = off / 27;
    const int c  = t2 & 7;          // channel within group
    const int s  = t2 >> 3;         // 0..63 (pos then neg, contiguous)

    const int center = centers[row * 64 + s];
    const int z = center >> 12, y = (center >> 6) & 63, x = center & 63;

    const int dz = o / 9 - 1;
    const int dy = (o / 3) % 3 - 1;
    const int dx = o % 3 - 1;
    const int zz = hp_clamp63(z + dz);   // replicate padding == clamp
    const int yy = hp_clamp63(y + dy);
    const int xx = hp_clamp63(x + dx);

    const size_t fidx = (((size_t)b * 64 + (size_t)(gidx * 8 + c)) * S_VOL) +
                        (size_t)zz * 4096 + (size_t)yy * 64 + (size_t)xx;
    out[i] = fmap[fidx];
}

extern "C" void kernel_launch(void* const* d_in, const int* in_sizes, int n_in,
                              void* d_out, int out_size, void* d_ws, size_t ws_size,
                              hipStream_t stream)
{
    (void)in_sizes; (void)n_in; (void)out_size; (void)ws_size;

    const float* fmap = (const float*)d_in[0];     // (2,64,64,64,64) f32
    const int*   mask = (const int*)d_in[1];       // (2,64,64,64) i32
    float*       out  = (float*)d_out;             // (2,8,1,13824) f32

    unsigned char* flags   = (unsigned char*)d_ws;                 // 2*262144 B
    int*           centers = (int*)((char*)d_ws + 2 * S_VOL);      // 16*64 ints

    hp_flags_kernel<<<1024, 512, 0, stream>>>(mask, flags);
    hp_sample_kernel<<<NROWS, 64, 0, stream>>>(flags, centers);
    hp_gather_kernel<<<(OUT_N + 255) / 256, 256, 0, stream>>>(fmap, centers, out);
}